// SHSidechainDecoder_74191265071637
// MI455X (gfx1250) — compile-verified
//
#include <hip/hip_runtime.h>
#include <cstdint>
#include <cstddef>

// ---------------------------------------------------------------------------
// Problem constants (from the reference: NX=41, voxel grid 41^3, C=4, K=4)
// ---------------------------------------------------------------------------
#define GVOX   68921          // 41^3 voxels per channel
#define NQ     17231          // ceil(GVOX/4) float4 quads (last quad padded)
#define SPAD   68924          // LDS floats incl. 3 pad slots
#define MASKW  2154           // ceil(SPAD/32) bitmask words
#define CCH    4              // channels per node
#define KPK    4              // peaks per channel
#define NEGF   (-1.0e9f)
#define THRESHF (-1.0e8f)

typedef unsigned int v4u __attribute__((ext_vector_type(4)));
typedef int          v8i __attribute__((ext_vector_type(8)));
typedef int          v4i __attribute__((ext_vector_type(4)));

// ---------------------------------------------------------------------------
// Kernel A: build sphere-mask bitmask (bit g set iff voxel g inside r<=6.0).
// Uses the harness-provided grid_xyz so the fp32 norm matches numpy exactly.
// ---------------------------------------------------------------------------
__global__ void sphere_mask_kernel(const float* __restrict__ gxyz,
                                   unsigned int* __restrict__ bits) {
  int w = blockIdx.x * blockDim.x + threadIdx.x;
  if (w >= MASKW) return;
  unsigned int word = 0u;
  for (int b = 0; b < 32; ++b) {
    int g = (w << 5) + b;
    if (g < GVOX) {
      float x = gxyz[3 * g + 0];
      float y = gxyz[3 * g + 1];
      float z = gxyz[3 * g + 2];
      // ((x*x + y*y) + z*z) with no fma-contraction, then IEEE sqrt -> matches np
      float s = __fadd_rn(__fadd_rn(__fmul_rn(x, x), __fmul_rn(y, y)),
                          __fmul_rn(z, z));
      if (sqrtf(s) <= 6.0f) word |= (1u << b);
    }
  }
  bits[w] = word;
}

// ---------------------------------------------------------------------------
// Kernel B: one workgroup per (n,c) channel.
//   - TDM: tensor_load_to_lds pulls the whole 269 KB channel HBM -> LDS once.
//   - burn sphere mask into LDS as NEG.
//   - K=4 rounds: value-only LDS scan (max3 chains), equality rescan for the
//     argmax index (first-occurrence via atomicMin), O(343) cube suppression.
// ---------------------------------------------------------------------------
__global__ __launch_bounds__(1024, 1) void nms_peaks_kernel(
    const float* __restrict__ density,
    const float* __restrict__ gxyz,
    const float* __restrict__ Rm,     // [N,3,3]
    const float* __restrict__ tp,     // [N,3]
    const float* __restrict__ nmask,  // [N]
    const unsigned int* __restrict__ bits,
    float* __restrict__ out, int NC)
{
  __shared__ alignas(16) float sv[SPAD];   // 275,696 B channel copy
  __shared__ float red[1024];
  __shared__ float sM;
  __shared__ unsigned int sG;

  const int tid = threadIdx.x;
  const int ch  = blockIdx.x;                 // 0 .. NC-1   (ch = n*C + c)
  const float* src = density + (size_t)ch * GVOX;

#if defined(__gfx1250__) && __has_builtin(__builtin_amdgcn_tensor_load_to_lds) && \
    __has_builtin(__builtin_amdgcn_s_wait_tensorcnt)
  if (tid == 0) {
    unsigned long long ga = (unsigned long long)(uintptr_t)src;       // byte addr
    unsigned int lds = (unsigned int)(uintptr_t)(&sv[0]);             // low32 = LDS offset
    // ---- D# group 0 (128b): count=1 | lds_addr | global_addr | type=2 ----
    v4u g0;
    g0[0] = 1u;                                                // count=1, user mode
    g0[1] = lds;                                               // lds_addr[63:32]
    g0[2] = (unsigned int)(ga & 0xFFFFFFFFull);                // global_addr[95:64]
    g0[3] = (unsigned int)((ga >> 32) & 0x1FFFFFFull)          // global_addr[120:96]
            | 0x80000000u;                                     // type=2 ("image")
    // ---- D# group 1 (256b): 2D tensor 1681 x 41, full-tile copy ----------
    v8i g1;
    g1[0] = 0x20000;                       // wg_mask=0, data_size=2 (4 bytes)
    g1[1] = (int)((1681u & 0xFFFFu) << 16);// tensor_dim0 lo16 @ [63:48]
    g1[2] = (int)(41u << 16);              // tensor_dim0 hi16=0 | tensor_dim1 lo16 @ [95:80]
    g1[3] = (int)(1681u << 16);            // tensor_dim1 hi16=0 | tile_dim0 @ [127:112]
    g1[4] = 41;                            // tile_dim1 @ [143:128], tile_dim2=0
    g1[5] = 1681;                          // tensor_dim0_stride[31:0]
    g1[6] = 0;                             // stride hi | tensor_dim1_stride lo (unused)
    g1[7] = 0;
    v4i g2 = (v4i){0, 0, 0, 0};            // dims 2/3 unused (2D)
    v4i g3 = (v4i){0, 0, 0, 0};
    v8i g4 = (v8i){0, 0, 0, 0, 0, 0, 0, 0};  // trailing group (clang-23 6-arg form)
    __builtin_amdgcn_tensor_load_to_lds(g0, g1, g2, g3, g4, 0);
    __builtin_amdgcn_s_wait_tensorcnt(0);
  }
#else
  // Fallback: plain coalesced copy HBM -> LDS
  for (int i = tid; i < GVOX; i += 1024) sv[i] = src[i];
#endif
  __syncthreads();

  // ---- burn sphere mask (and the 3 pad slots) into LDS as NEG -------------
  for (int i = tid; i < NQ; i += 1024) {
    int q = i << 2;
    unsigned int nib = (bits[q >> 5] >> (q & 31)) & 0xFu;
    if (nib != 0xFu) {
      if (!(nib & 1u)) sv[q + 0] = NEGF;
      if (!(nib & 2u)) sv[q + 1] = NEGF;
      if (!(nib & 4u)) sv[q + 2] = NEGF;
      if (!(nib & 8u)) sv[q + 3] = NEGF;
    }
  }

  // ---- per-node frame (only thread 0 uses these) --------------------------
  const int nn = ch >> 2;   // C = 4
  float R00=0,R01=0,R02=0,R10=0,R11=0,R12=0,R20=0,R21=0,R22=0,t0=0,t1=0,t2=0,m=0;
  if (tid == 0) {
    const float* R = Rm + (size_t)nn * 9;
    R00=R[0]; R01=R[1]; R02=R[2]; R10=R[3]; R11=R[4]; R12=R[5];
    R20=R[6]; R21=R[7]; R22=R[8];
    t0 = tp[nn*3+0]; t1 = tp[nn*3+1]; t2 = tp[nn*3+2];
    m  = nmask[nn];
  }

  const int cgOff = NC * 12;   // coords_global offset
  const int scOff = NC * 24;   // scores offset
  const int mkOff = NC * 28;   // mask offset

  for (int k = 0; k < KPK; ++k) {
    __syncthreads();                       // orders prior suppression/mask writes
    // ---- value-only scan: ds_load_b128 + max3 chain ----
    float tmax = -3.402823466e38f;
    for (int i = tid; i < NQ; i += 1024) {
      const float4 v = *(const float4*)&sv[i << 2];
      tmax = fmaxf(tmax, fmaxf(fmaxf(v.x, v.y), fmaxf(v.z, v.w)));
    }
    red[tid] = tmax;
    __syncthreads();
    if (tid == 0) sG = 0xFFFFFFFFu;
    if (tid < 32) {
      float v = red[tid];
      for (int j = tid + 32; j < 1024; j += 32) v = fmaxf(v, red[j]);
      for (int off = 16; off > 0; off >>= 1) v = fmaxf(v, __shfl_xor(v, off, 32));
      if (tid == 0) sM = v;
    }
    __syncthreads();
    const float M = sM;

    // ---- index recovery: only matching threads rescan; first index wins ----
    if (tmax == M) {
      for (int i = tid; i < NQ; i += 1024) {
        const float4 v = *(const float4*)&sv[i << 2];
        if (v.x == M || v.y == M || v.z == M || v.w == M) {
          int off = (v.x == M) ? 0 : (v.y == M) ? 1 : (v.z == M) ? 2 : 3;
          atomicMin(&sG, (unsigned int)((i << 2) + off));
          break;
        }
      }
    }
    __syncthreads();
    unsigned int g = sG;
    if (g == 0xFFFFFFFFu) g = 0u;
    const bool valid = (M > THRESHF);
    const int gi = (int)g;
    const int x0 = gi / 1681;
    const int rr = gi - x0 * 1681;
    const int y0 = rr / 41;
    const int z0 = rr - y0 * 41;

    // ---- emit outputs for this (ch, k) ----
    if (tid == 0) {
      unsigned int gc = (g < (unsigned)GVOX) ? g : 0u;   // clamp pad idx (invalid only)
      float plx = 0.f, ply = 0.f, plz = 0.f;
      if (valid) { plx = gxyz[3*gc]; ply = gxyz[3*gc+1]; plz = gxyz[3*gc+2]; }
      const float pgx = (R00*plx + R01*ply + R02*plz + t0) * m;
      const float pgy = (R10*plx + R11*ply + R12*plz + t1) * m;
      const float pgz = (R20*plx + R21*ply + R22*plz + t2) * m;
      const int base = ch * KPK + k;
      out[base*3 + 0] = plx * m;
      out[base*3 + 1] = ply * m;
      out[base*3 + 2] = plz * m;
      out[cgOff + base*3 + 0] = pgx;
      out[cgOff + base*3 + 1] = pgy;
      out[cgOff + base*3 + 2] = pgz;
      out[scOff + base] = (valid ? M : NEGF) * m;
      out[mkOff + base] = (valid && (m != 0.f)) ? 1.f : 0.f;
    }

    // ---- NMS suppression: 7x7x7 Chebyshev cube, direct LDS writes ----
    if (valid && k < (KPK - 1) && tid < 343) {
      const int dz = (tid % 7) - 3;
      const int dy = ((tid / 7) % 7) - 3;
      const int dx = (tid / 49) - 3;
      const int xx = x0 + dx, yy = y0 + dy, zz = z0 + dz;
      if ((unsigned)xx < 41u && (unsigned)yy < 41u && (unsigned)zz < 41u)
        sv[(xx * 41 + yy) * 41 + zz] = NEGF;
    }
  }
}

// ---------------------------------------------------------------------------
// Inputs (setup_inputs order):
//  0 density [1,128,4,68921] f32 | 1 grid_xyz [G,3] f32 | 2 sphere_mask (unused)
//  3 coords_int (unused) | 4 Rmats [1,128,3,3] | 5 tpos [1,128,3] | 6 node_mask [1,128]
// Output: coords_local(6144) | coords_global(6144) | scores(2048) | mask(2048)
// ---------------------------------------------------------------------------
extern "C" void kernel_launch(void* const* d_in, const int* in_sizes, int n_in,
                              void* d_out, int out_size, void* d_ws, size_t ws_size,
                              hipStream_t stream) {
  const float* density  = (const float*)d_in[0];
  const float* grid_xyz = (const float*)d_in[1];
  const float* Rmats    = (const float*)d_in[4];
  const float* tpos     = (const float*)d_in[5];
  const float* nodemask = (const float*)d_in[6];
  float* out = (float*)d_out;

  unsigned int* bits = (unsigned int*)d_ws;   // needs MASKW*4 = 8,616 bytes

  const int NC = in_sizes[0] / GVOX;          // 512 channels (N*C)

  sphere_mask_kernel<<<(MASKW + 255) / 256, 256, 0, stream>>>(grid_xyz, bits);
  nms_peaks_kernel<<<NC, 1024, 0, stream>>>(density, grid_xyz, Rmats, tpos,
                                            nodemask, bits, out, NC);
}